// HaarLayer_6889127542790
// MI455X (gfx1250) — compile-verified
//
#include <hip/hip_runtime.h>

// Haar 3-axis butterfly (C,H,W) for x:(32,32,256,256) f32 -> out:(32,8,16,128,128) f32.
// out[n,s,c,h,w] = sum_{i,j,k} (-1)^(bC*i + bH*j + bW*k) * x[n,2c+i,2h+j,2w+k],
// with s = 4*bW + 2*bH + bC.  Implemented as D = A(Hadamard) x B(data) via
// four chained V_WMMA_F32_16X16X4_F32 (K = 8 combos x 2 (c,h)-groups).

typedef __attribute__((ext_vector_type(2))) float v2f;
typedef __attribute__((ext_vector_type(8))) float v8f;

#define N_DIM   32
#define C_IN    32
#define H_IN    256
#define W_IN    256
// out dims: (32, 8, 16, 128, 128)
#define OUT_S_STRIDE (16L * 128 * 128)   // 262144 elements between sub-bands

__global__ __launch_bounds__(256) void haar_wmma_kernel(const float* __restrict__ in,
                                                        float* __restrict__ out) {
  const int lane = threadIdx.x & 31;
  const int wave = blockIdx.x * (blockDim.x >> 5) + (threadIdx.x >> 5);
  // tile decode: 8 w-tiles x 64 h-tiles x 16 c x 32 n = 262144 wave-tiles
  const int wt = wave & 7;          // 16 output w per tile
  const int ht = (wave >> 3) & 63;  // 2 output h rows per tile (groups A/B)
  const int c  = (wave >> 9) & 15;
  const int n  =  wave >> 13;
  if (n >= N_DIM) return;           // uniform per wave; grid is exact

  const int half = lane >> 4;       // B: selects input row j; D: selects h-group
  const int p    = lane & 15;       // w position within tile

  // ---------- loads (B matrices) ----------
  // chunk (i,g): channel 2c+i, input row 4*ht + 2g + half, W pair at 2*(wt*16+p)
  // B layout (4x16, 2 VGPRs): lanes0-15 -> K'=0(V0),1(V1) = row j=0, k=0/1
  //                           lanes16-31 -> K'=2,3        = row j=1, k=0/1
  const long idx0 = ((long)(n * C_IN + 2 * c) * H_IN + (4 * ht + half)) * (long)W_IN
                    + 2L * (wt * 16 + p);
  const float* base = in + idx0;
  v2f b0 = __builtin_nontemporal_load((const v2f*)(base));                 // grpA, i=0
  v2f b1 = __builtin_nontemporal_load((const v2f*)(base + 65536));         // grpA, i=1 (ch+1)
  v2f b2 = __builtin_nontemporal_load((const v2f*)(base + 512));           // grpB, i=0 (+2 rows)
  v2f b3 = __builtin_nontemporal_load((const v2f*)(base + 65536 + 512));   // grpB, i=1

  // ---------- A (block-diagonal Hadamard), per-lane ----------
  // A layout (16x4, 2 VGPRs): lane M = lane%16 = output row (sub-band slot),
  //   lanes0-15: V0=K0 (j=0,k=0), V1=K1 (j=0,k=1); lanes16-31: V0=K2 (j=1,k=0), V1=K3.
  // Rows 0-7 = group A sub-bands, rows 8-15 = group B sub-bands (block diagonal).
  const int M  = lane & 15;
  const int s  = M & 7;
  const int bC = s & 1, bH = (s >> 1) & 1, bW = (s >> 2) & 1;
  const float e  = (bH & half)        ? -1.0f : 1.0f;   // k=0, i=0 sign
  const float eb = ((bH & half) ^ bW) ? -1.0f : 1.0f;   // k=1, i=0 sign
  const float sC = bC ? -1.0f : 1.0f;                   // extra factor for i=1
  const float mA = (M < 8) ? 1.0f : 0.0f;               // group-A rows live in M<8
  const float mB = 1.0f - mA;                           // group-B rows in M>=8
  v2f a0 = { mA * e,      mA * eb      };               // K 0-3 : grpA, i=0
  v2f a1 = { mA * e * sC, mA * eb * sC };               // K 4-7 : grpA, i=1
  v2f a2 = { mB * e,      mB * eb      };               // K 8-11: grpB, i=0
  v2f a3 = { mB * e * sC, mB * eb * sC };               // K12-15: grpB, i=1

  // ---------- four chained f32 WMMAs: D = sum A_q * B_q ----------
  v8f d = {};
  d = __builtin_amdgcn_wmma_f32_16x16x4_f32(false, a0, false, b0, (short)0, d, false, false);
  d = __builtin_amdgcn_wmma_f32_16x16x4_f32(false, a1, false, b1, (short)0, d, false, false);
  d = __builtin_amdgcn_wmma_f32_16x16x4_f32(false, a2, false, b2, (short)0, d, false, false);
  d = __builtin_amdgcn_wmma_f32_16x16x4_f32(false, a3, false, b3, (short)0, d, false, false);

  // ---------- stores ----------
  // D VGPR r, lane: row = r + half*8 -> (h-group = half, sub-band = r), col = p.
  // out[n, r, c, 2*ht + half, wt*16 + p]; sub-band stride folds into the 24-bit imm.
  const long ob = ((long)n * 8 * 16 + c) * (128L * 128)
                  + (long)(2 * ht + half) * 128 + (wt * 16 + p);
  float* o = out + ob;
  __builtin_nontemporal_store(d[0], o + 0 * OUT_S_STRIDE);
  __builtin_nontemporal_store(d[1], o + 1 * OUT_S_STRIDE);
  __builtin_nontemporal_store(d[2], o + 2 * OUT_S_STRIDE);
  __builtin_nontemporal_store(d[3], o + 3 * OUT_S_STRIDE);
  __builtin_nontemporal_store(d[4], o + 4 * OUT_S_STRIDE);
  __builtin_nontemporal_store(d[5], o + 5 * OUT_S_STRIDE);
  __builtin_nontemporal_store(d[6], o + 6 * OUT_S_STRIDE);
  __builtin_nontemporal_store(d[7], o + 7 * OUT_S_STRIDE);
}

extern "C" void kernel_launch(void* const* d_in, const int* in_sizes, int n_in,
                              void* d_out, int out_size, void* d_ws, size_t ws_size,
                              hipStream_t stream) {
  (void)in_sizes; (void)n_in; (void)out_size; (void)d_ws; (void)ws_size;
  const float* x = (const float*)d_in[0];
  float* out = (float*)d_out;
  // 262144 wave-tiles / 8 waves per 256-thread block = 32768 blocks
  haar_wmma_kernel<<<32768, 256, 0, stream>>>(x, out);
}